// Spherical_Voxelization_69020124447043
// MI455X (gfx1250) — compile-verified
//
#include <hip/hip_runtime.h>
#include <math.h>

// Spherical voxelization: 16 x 65536 x 3 fp32 points -> 16 x (5*30*15) histogram.
// VALU-bound (sqrt/atan2/acos per point). Closed-form two-nearest-bin per axis
// (branchless; replaces reference's 50-distance top_k), privatized LDS histogram
// per block (ds_add_f32, wave32), global_atomic_add_f32 flush.

#define RAD_N 5
#define AZI_N 30
#define ELE_N 15
#define HIST_BINS (RAD_N * AZI_N * ELE_N) // 2250
#define DES_R 2.0f
#define TWO_PI_F 6.28318530717958647692f
#define PI_F 3.14159265358979323846f

__global__ void sphvox_zero_kernel(float* __restrict__ out, int n) {
  int i = blockIdx.x * blockDim.x + threadIdx.x;
  if (i < n) out[i] = 0.0f;
}

__global__ __launch_bounds__(256) void sphvox_hist_kernel(
    const float* __restrict__ pts,   // [B, N, 3]
    float* __restrict__ out,         // [B, HIST_BINS]
    int N, int pts_per_block) {
  __shared__ float hist[HIST_BINS];
  for (int i = threadIdx.x; i < HIST_BINS; i += blockDim.x) hist[i] = 0.0f;
  __syncthreads();

  const int batch = blockIdx.y;
  const int base  = blockIdx.x * pts_per_block;
  const int pend  = min(base + pts_per_block, N);
  const float* __restrict__ bpts = pts + (size_t)batch * (size_t)N * 3u;

  const float tr     = DES_R / (float)RAD_N;   // 0.4
  const float inv_tr = (float)RAD_N / DES_R;
  const float sr     = 0.5f * tr;
  const float inv_ta = (float)AZI_N / TWO_PI_F;
  const float te     = PI_F / (float)ELE_N;
  const float inv_te = (float)ELE_N / PI_F;
  const float se     = 0.5f * te;

  for (int p = base + (int)threadIdx.x; p < pend; p += (int)blockDim.x) {
    // Keep the streaming read ahead of the ALU-bound body (global_prefetch_b8).
    __builtin_prefetch(&bpts[3 * (p + 8 * (int)blockDim.x)], 0, 1);

    const float x = bpts[3 * p + 0];
    const float y = bpts[3 * p + 1];
    const float z = bpts[3 * p + 2];

    const float r = sqrtf(x * x + y * y + z * z);
    if (r > DES_R) continue;  // reference: wr = 0 outside sphere -> no contribution

    float azi = atan2f(y, x);
    azi = (azi < 0.0f) ? azi + TWO_PI_F : azi;             // [0, 2pi]
    const float cz  = fminf(fmaxf(z / fmaxf(r, 1e-12f), -1.0f), 1.0f);
    const float ele = acosf(cz);                           // [0, pi]

    // ---- radial axis (branchless; half-bin clamps per reference) ----
    int ir0, ir1; float wr0, wr1;
    {
      const float fr = r * inv_tr;                  // [0, 5]
      int j = (int)floorf(fr); j = min(j, RAD_N - 1);
      const float u  = fr - (float)j;               // [0, 1]
      const float d0 = fabsf(u - 0.5f);             // distance in bin units
      const bool low  = r < sr;                     // j == 0 here
      const bool high = r > (DES_R - sr);           // j == RAD_N-1 here
      int jn = (u >= 0.5f) ? j + 1 : j - 1;         // OOR only when d0 == 0 (w1==0)
      jn = max(0, min(RAD_N - 1, jn));
      ir0 = j;
      wr0 = low ? 1.0f : (1.0f - d0);
      ir1 = low ? 1 : (high ? (RAD_N - 2) : jn);
      wr1 = (low | high) ? 0.0f : d0;
    }

    // ---- azimuth axis (circular, branchless) ----
    int ia0, ia1; float wa0, wa1;
    {
      const float fa = azi * inv_ta;                // [0, 30]
      const int j = (int)floorf(fa);
      const float u  = fa - (float)j;
      const float d0 = fabsf(u - 0.5f);
      int i0 = j;                        i0 = (i0 >= AZI_N) ? i0 - AZI_N : i0;
      int jn = (u >= 0.5f) ? j + 1 : j - 1;
      jn = (jn >= AZI_N) ? jn - AZI_N : jn;
      jn = (jn < 0) ? jn + AZI_N : jn;
      ia0 = i0; wa0 = 1.0f - d0;
      ia1 = jn; wa1 = d0;
    }

    // ---- elevation axis (branchless; both boundary clamps give w=(1,0)) ----
    int ie0, ie1; float we0, we1;
    {
      const float fe = ele * inv_te;                // [0, 15]
      int j = (int)floorf(fe); j = min(j, ELE_N - 1);
      const float u  = fe - (float)j;
      const float d0 = fabsf(u - 0.5f);
      const bool lo = ele < se;                     // j == 0 here
      const bool hi = ele > (PI_F - se);            // j == ELE_N-1 here
      int jn = (u >= 0.5f) ? j + 1 : j - 1;
      jn = max(0, min(ELE_N - 1, jn));
      ie0 = j;
      we0 = (lo | hi) ? 1.0f : (1.0f - d0);
      ie1 = lo ? 1 : (hi ? (ELE_N - 2) : jn);
      we1 = (lo | hi) ? 0.0f : d0;
    }

    // ---- 8 trilinear contributions -> LDS privatized histogram (ds_add_f32) ----
    const int   iR[2] = { ir0, ir1 }; const float wR[2] = { wr0, wr1 };
    const int   iA[2] = { ia0, ia1 }; const float wA[2] = { wa0, wa1 };
    const int   iE[2] = { ie0, ie1 }; const float wE[2] = { we0, we1 };
#pragma unroll
    for (int a = 0; a < 2; ++a) {
      const float wra = wR[a]; const int ba = iR[a] * AZI_N;
#pragma unroll
      for (int b = 0; b < 2; ++b) {
        const float wab = wra * wA[b]; const int bb = (ba + iA[b]) * ELE_N;
#pragma unroll
        for (int c = 0; c < 2; ++c) {
          atomicAdd(&hist[bb + iE[c]], wab * wE[c]);
        }
      }
    }
  }

  __syncthreads();

  // ---- flush block-private histogram to global (64 blocks/batch contend) ----
  float* __restrict__ bout = out + (size_t)batch * HIST_BINS;
  for (int i = threadIdx.x; i < HIST_BINS; i += blockDim.x) {
    const float v = hist[i];
    if (v != 0.0f) atomicAdd(&bout[i], v);
  }
}

extern "C" void kernel_launch(void* const* d_in, const int* in_sizes, int n_in,
                              void* d_out, int out_size, void* d_ws, size_t ws_size,
                              hipStream_t stream) {
  (void)n_in; (void)d_ws; (void)ws_size;
  const float* pts = (const float*)d_in[0];
  float* out = (float*)d_out;

  const int B = out_size / HIST_BINS;            // 16
  const int N = in_sizes[0] / (3 * B);           // 65536

  // Harness poisons d_out; the histogram accumulates via atomics, so zero first.
  sphvox_zero_kernel<<<(out_size + 255) / 256, 256, 0, stream>>>(out, out_size);

  // 1024 pts/block -> 1024 blocks total: ~4 blocks/WGP, 8 waves/SIMD32 to hide
  // the atan2/acos dependency chains; flush is ≤2250 atomics/block vs ≤8192 LDS atomics.
  const int pts_per_block = 1024;
  const int bx = (N + pts_per_block - 1) / pts_per_block;
  dim3 grid((unsigned)bx, (unsigned)B);
  sphvox_hist_kernel<<<grid, 256, 0, stream>>>(pts, out, N, pts_per_block);
}